// GNN_71296457114107
// MI455X (gfx1250) — compile-verified
//
#include <hip/hip_runtime.h>
#include <hip/hip_bf16.h>

typedef __attribute__((ext_vector_type(2))) float v2f;
typedef __attribute__((ext_vector_type(8))) float v8f;

#define GNN_N_NODES 100000
#define GEMM_WAVES 8

// ---------------------------------------------------------------------------
// Scatter-add message aggregation: agg[dst] += feat[src], float4-vectorized.
// Random accesses hit L2 (operands << 192 MB); atomics are hw f32 adds in L2.
// ---------------------------------------------------------------------------
__global__ void gnn_scatter_add_f4(const float* __restrict__ feat,
                                   const long long* __restrict__ src,
                                   const long long* __restrict__ dst,
                                   float* __restrict__ agg,
                                   int E, int F4, int F) {
    int t = blockIdx.x * blockDim.x + threadIdx.x;
    int total = E * F4;
    if (t >= total) return;
    int e = t / F4;
    int g = t - e * F4;
    long long s = src[e];
    long long d = dst[e];
    const float4 v = *(const float4*)(feat + (size_t)s * F + g * 4);
    float* out = agg + (size_t)d * F + g * 4;
    unsafeAtomicAdd(out + 0, v.x);
    unsafeAtomicAdd(out + 1, v.y);
    unsafeAtomicAdd(out + 2, v.z);
    unsafeAtomicAdd(out + 3, v.w);
}

// ---------------------------------------------------------------------------
// Fused GEMM: out = act( [a0 | a1] @ [wTop ; wBot] + bias )
// A is the row-wise concat of a0 (width WA0) and a1 (width WA1); W is the
// row-stack of wTop (KTOP rows) and wBot. WA1==0 => plain GEMM (fc layers).
// One wave32 computes a 16-row tile via v_wmma_f32_16x16x4_f32 chains.
//
// Fragment layout (ISA 7.12.2, wave32):
//   half = lane/16, m = lane%16
//   A vgpr r : A[row + m][k0 + r + 2*half]
//   B vgpr r : W[k0 + r + 2*half][nb + m]
//   D vgpr r : out[row + r + 8*half][nb + m]
// ---------------------------------------------------------------------------
template <int K, int NOUT, int WA0, int WA1, int KTOP, int RELU>
__global__ void gnn_gemm_wmma(const float* __restrict__ a0,
                              const float* __restrict__ a1,
                              const float* __restrict__ wTop,
                              const float* __restrict__ wBot,
                              const float* __restrict__ bias,
                              float* __restrict__ out,
                              int nrows) {
    __shared__ float sW[K * NOUT];
    __shared__ float sB[NOUT];

    // Stage fused weight matrix + bias in LDS (once per block).
    for (int i = threadIdx.x; i < K * NOUT; i += blockDim.x) {
        int k = i / NOUT, n = i - k * NOUT;
        float w;
        if (k < KTOP) w = wTop[k * NOUT + n];
        else          w = wBot[(k - KTOP) * NOUT + n];
        sW[i] = w;
    }
    for (int i = threadIdx.x; i < NOUT; i += blockDim.x) sB[i] = bias[i];
    __syncthreads();

    const int wave = threadIdx.x >> 5;
    const int lane = threadIdx.x & 31;
    const int half = lane >> 4;   // 0: lanes 0-15, 1: lanes 16-31
    const int m    = lane & 15;

    const int tile = blockIdx.x * GEMM_WAVES + wave;
    const int row  = tile * 16;
    if (row >= nrows) return;     // wave-uniform: EXEC stays all-ones below

    #pragma unroll
    for (int nb = 0; nb < NOUT; nb += 16) {
        v8f acc = {};
        #pragma unroll
        for (int k0 = 0; k0 < K; k0 += 4) {
            const int ka = k0 + 2 * half;        // ka, ka+1 in [k0, k0+3]
            const float* pa;
            if (k0 < WA0) {                      // compile-time after unroll
                pa = a0 + (size_t)(row + m) * WA0 + ka;
            } else {
                pa = a1 + (size_t)(row + m) * WA1 + (ka - WA0);
            }
            v2f av = { pa[0], pa[1] };
            v2f bv = { sW[ka * NOUT + nb + m], sW[(ka + 1) * NOUT + nb + m] };
            acc = __builtin_amdgcn_wmma_f32_16x16x4_f32(
                /*neg_a=*/false, av, /*neg_b=*/false, bv,
                /*c_mod=*/(short)0, acc, /*reuse_a=*/false, /*reuse_b=*/false);
        }
        const float bval = sB[nb + m];
        #pragma unroll
        for (int r = 0; r < 8; ++r) {
            float v = acc[r] + bval;
            if (RELU) v = fmaxf(v, 0.0f);
            out[(size_t)(row + r + 8 * half) * NOUT + nb + m] = v;
        }
    }
}

// ---------------------------------------------------------------------------
extern "C" void kernel_launch(void* const* d_in, const int* in_sizes, int n_in,
                              void* d_out, int out_size, void* d_ws, size_t ws_size,
                              hipStream_t stream) {
    const float*     x       = (const float*)d_in[0];
    const long long* ei      = (const long long*)d_in[1];   // int64 [2, E]
    const float*     w_root1 = (const float*)d_in[2];
    const float*     w_rel1  = (const float*)d_in[3];
    const float*     b1      = (const float*)d_in[4];
    const float*     w_root2 = (const float*)d_in[5];
    const float*     w_rel2  = (const float*)d_in[6];
    const float*     b2      = (const float*)d_in[7];
    const float*     fc1_w   = (const float*)d_in[8];
    const float*     fc1_b   = (const float*)d_in[9];
    const float*     fc2_w   = (const float*)d_in[10];
    const float*     fc2_b   = (const float*)d_in[11];
    float*           out     = (float*)d_out;

    const int N = GNN_N_NODES;
    const int E = in_sizes[1] / 2;                          // 1,600,000

    // Workspace layout (floats)
    float* agg1 = (float*)d_ws;                  // [N,16]
    float* agg2 = agg1 + (size_t)N * 16;         // [N,32]
    float* h1   = agg2 + (size_t)N * 32;         // [N,32]
    float* h2   = h1   + (size_t)N * 32;         // [N,64]
    float* h3   = h2   + (size_t)N * 64;         // [N,32]

    hipMemsetAsync(agg1, 0, (size_t)N * 16 * sizeof(float), stream);
    hipMemsetAsync(agg2, 0, (size_t)N * 32 * sizeof(float), stream);

    const long long* srcI = ei;
    const long long* dstI = ei + E;

    const int tiles  = (N + 15) / 16;                       // 6250
    const int blocks = (tiles + GEMM_WAVES - 1) / GEMM_WAVES;
    const int bthr   = GEMM_WAVES * 32;

    // Layer 1 aggregation: agg1 += x[src]
    {
        int total = E * 4, bs = 256;
        gnn_scatter_add_f4<<<(total + bs - 1) / bs, bs, 0, stream>>>(
            x, srcI, dstI, agg1, E, 4, 16);
    }
    // h1 = relu([agg1 | x] @ [w_rel1 ; w_root1] + b1)   (K=32, Nout=32)
    gnn_gemm_wmma<32, 32, 16, 16, 16, 1><<<blocks, bthr, 0, stream>>>(
        agg1, x, w_rel1, w_root1, b1, h1, N);

    // Layer 2 aggregation: agg2 += h1[src]
    {
        int total = E * 8, bs = 256;
        gnn_scatter_add_f4<<<(total + bs - 1) / bs, bs, 0, stream>>>(
            h1, srcI, dstI, agg2, E, 8, 32);
    }
    // h2 = relu([agg2 | h1] @ [w_rel2 ; w_root2] + b2)  (K=64, Nout=64)
    gnn_gemm_wmma<64, 64, 32, 32, 32, 1><<<blocks, bthr, 0, stream>>>(
        agg2, h1, w_rel2, w_root2, b2, h2, N);

    // h3 = relu(h2 @ fc1_w + fc1_b)                     (K=64, Nout=32)
    gnn_gemm_wmma<64, 32, 64, 0, 64, 1><<<blocks, bthr, 0, stream>>>(
        h2, (const float*)nullptr, fc1_w, (const float*)nullptr, fc1_b, h3, N);

    // out = h3 @ fc2_w + fc2_b                          (K=32, Nout=16)
    gnn_gemm_wmma<32, 16, 32, 0, 32, 0><<<blocks, bthr, 0, stream>>>(
        h3, (const float*)nullptr, fc2_w, (const float*)nullptr, fc2_b, out, N);
}